// ConvTranspose4d_11991548690493
// MI455X (gfx1250) — compile-verified
//
#include <hip/hip_runtime.h>
#include <hip/hip_bf16.h>

// Problem constants (from reference)
#define Cc   8
#define Tt   8
#define Dd   24
#define Hh   48
#define Ww   48
#define KT   3
#define KS   3
#define TO   6      // T - (KT-1)
#define DO   47     // (D-1)*2 - 2*1 + 3
#define HO   95
#define WO   95
#define WT   6      // ceil(WO/16)
#define NTAP 81     // 3*3*3*3
#define XCH  (Tt*Dd*Hh*Ww)   // per-channel x stride = 442368
#define XROW (Hh*Ww)         // 2304

typedef __attribute__((ext_vector_type(2))) float v2f;
typedef __attribute__((ext_vector_type(8))) float v8f;

// One wave computes a 16(co-padded) x 16(ow) fp32 tile for a fixed (f, od, oh)
// via V_WMMA_F32_16X16X4_F32, K = 4 input channels per step (2 steps cover ci=8),
// accumulating over the (kt,kd,kh,kw) taps of the dilated forward conv that is
// equivalent to the stride-2 transposed conv. Gathers are branchless: per
// (kt,kd,kh) each lane needs only the pair {x[ib], x[ib+1]}; the kw selection
// is applied with precomputed {0,1} float masks (mul+fma, no compares in loop).
__global__ __launch_bounds__(128)
void convT4d_wmma_kernel(const float* __restrict__ x,
                         const float* __restrict__ weight,
                         float* __restrict__ out) {
    // ---- Stage flipped+transposed weights into LDS: wlds[tap][co16][ci8] ----
    // Wf[co,ci,i,kd,kh,kw] = weight[ci,co,i,2-kd,2-kh,2-kw]; co padded 8->16 with 0.
    // Layout puts a lane's two consecutive K (=ci) values adjacent -> ds_load_b64.
    __shared__ float wlds[NTAP * 16 * Cc];   // 10368 floats = 41.5 KB

    for (int idx = threadIdx.x; idx < NTAP * 16 * Cc; idx += blockDim.x) {
        int ci  = idx & 7;
        int co  = (idx >> 3) & 15;
        int tap = idx >> 7;
        int kw  = tap % 3;
        int kh  = (tap / 3) % 3;
        int kd  = (tap / 9) % 3;
        int i   = tap / 27;
        float v = 0.0f;
        if (co < Cc) {
            int widx = ((((ci * Cc + co) * KT + i) * KS + (2 - kd)) * KS + (2 - kh)) * KS + (2 - kw);
            v = weight[widx];
        }
        wlds[idx] = v;
    }
    __syncthreads();

    // ---- Per-wave tile ----
    const int wave = threadIdx.x >> 5;
    const int lane = threadIdx.x & 31;
    const int tid  = blockIdx.x * 4 + wave;          // wave-uniform tile id
    const int total = TO * DO * HO * WT;
    if (tid >= total) return;                        // wave-uniform exit

    int t   = tid;
    const int wt  = t % WT;  t /= WT;
    const int oh  = t % HO;  t /= HO;
    const int od  = t % DO;  t /= DO;
    const int f   = t;                               // 0..5
    const int owb = wt * 16;

    const int n  = lane & 15;        // WMMA column (ow offset) / A row (co)
    const int k0 = (lane >> 4) * 2;  // base K index for this half-wave

    // Base of the per-lane x pair: all kw taps read x[ib] or x[ib+1].
    const int ib0 = (owb + n - 1) >> 1;
    const int ib  = ib0 < 0 ? 0 : (ib0 > (Ww - 2) ? (Ww - 2) : ib0);

    // Loop-invariant {0,1} selection masks per kw tap (valid => iw in {ib, ib+1}).
    float mlo[3], mhi[3];
    #pragma unroll
    for (int kw = 0; kw < KS; ++kw) {
        const int  mw    = owb + n + kw - 1;
        const int  iw    = mw >> 1;
        const bool valid = ((mw & 1) == 0) && (mw >= 0) && (iw < Ww);
        mlo[kw] = (valid && iw == ib)     ? 1.0f : 0.0f;
        mhi[kw] = (valid && iw == ib + 1) ? 1.0f : 0.0f;
    }

    v8f acc = {};

    #pragma unroll
    for (int i = 0; i < KT; ++i) {
        const int frame = f + i;                     // always < 8
        #pragma unroll
        for (int kd = 0; kd < KS; ++kd) {
            const int md = od + kd - 1;              // wave-uniform
            if (md & 1) continue;                    // odd (incl. md=-1/47) -> skip
            const int id = md >> 1;                  // always in [0,23]
            #pragma unroll
            for (int kh = 0; kh < KS; ++kh) {
                const int mh = oh + kh - 1;          // wave-uniform
                if (mh & 1) continue;                // odd (incl. mh=-1/95) -> skip
                const int ih = mh >> 1;              // always in [0,47]

                const int rowbase = (frame * Dd + id) * XROW + ih * Ww;
                __builtin_prefetch(x + rowbase, 0, 0);   // global_prefetch_b8

                // Unconditional, in-bounds pair loads for this lane's 4 ci values.
                float plo[4], phi[4];
                #pragma unroll
                for (int c4 = 0; c4 < 4; ++c4) {
                    const int ci = (c4 >> 1) * 4 + k0 + (c4 & 1);
                    const float* p = x + ci * XCH + rowbase + ib;
                    plo[c4] = p[0];
                    phi[c4] = p[1];
                }

                #pragma unroll
                for (int kw = 0; kw < KS; ++kw) {
                    const int tapoff = (((i * 3 + kd) * 3 + kh) * 3 + kw) << 7;
                    #pragma unroll
                    for (int chunk = 0; chunk < 2; ++chunk) {
                        // A-fragment: two adjacent K values, 8B-aligned -> ds_load_b64
                        const v2f aa = *(const v2f*)&wlds[tapoff + n * Cc + chunk * 4 + k0];
                        v2f bb;
                        #pragma unroll
                        for (int v = 0; v < 2; ++v) {
                            const int c4 = chunk * 2 + v;
                            // mask-select from the pair: mul + fma, no compares
                            bb[v] = plo[c4] * mlo[kw] + phi[c4] * mhi[kw];
                        }
                        // D(16x16,f32) = A(16x4,f32) x B(4x16,f32) + C
                        acc = __builtin_amdgcn_wmma_f32_16x16x4_f32(
                                  false, aa, false, bb, (short)0, acc, false, false);
                    }
                }
            }
        }
    }

    // ---- Store: C/D layout -> VGPR r, lanes 0..15 hold M=r (co=r), N=lane ----
    if (lane < 16) {
        const int ow = owb + lane;
        if (ow < WO) {
            const int obase = ((f * DO + od) * HO + oh) * WO + ow;
            #pragma unroll
            for (int r = 0; r < Cc; ++r) {
                out[r * (TO * DO * HO * WO) + obase] = acc[r];
            }
        }
    }
}

extern "C" void kernel_launch(void* const* d_in, const int* in_sizes, int n_in,
                              void* d_out, int out_size, void* d_ws, size_t ws_size,
                              hipStream_t stream) {
    const float* x = (const float*)d_in[0];
    const float* w = (const float*)d_in[1];
    float* out = (float*)d_out;

    const int total  = TO * DO * HO * WT;   // wave-tiles
    const int blocks = (total + 3) / 4;     // 4 waves (128 threads) per block
    convT4d_wmma_kernel<<<blocks, 128, 0, stream>>>(x, w, out);
}